// EquivSetGNN_28226525069818
// MI455X (gfx1250) — compile-verified
//
#include <hip/hip_runtime.h>
#include <hip/hip_bf16.h>

// ---------------------------------------------------------------------------
// EquivSetGNN forward for MI455X (gfx1250, wave32, WMMA).
// - All dense matmuls use V_WMMA_F32_16X16X4_F32 (exact fp32).
// - Weights pre-packed into (k-pair, col) interleaved layout so B fragments
//   are single aligned ds_load_b64 ops (no VGPR shuffling).
// - W chunks staged into LDS with gfx1250 async loads
//   (global_load_async_to_lds_b128 + s_wait_asynccnt).
// - Each wave computes two 16-row tiles sharing B fragments (16 WMMA per
//   8 LDS loads).
// - The big [E,256]x[256,128] GEMM fuses gather + LayerNorm on the A side and
//   scatter-mean accumulation (fp32 global atomics) on the output side.
// ---------------------------------------------------------------------------

typedef float v2f __attribute__((ext_vector_type(2)));
typedef float v8f __attribute__((ext_vector_type(8)));

#define N_NODES  50000
#define N_HEDGES 100000
#define N_INC    600000
#define F_IN     512
#define HID      128
#define LNEPS    1e-5f
#define ALPHAC   0.5f

__device__ __forceinline__ v8f wmma_f32(v2f a, v2f b, v8f c) {
  // 8 args: (neg_a, A, neg_b, B, c_mod, C, reuse_a, reuse_b)
  return __builtin_amdgcn_wmma_f32_16x16x4_f32(false, a, false, b, (short)0, c,
                                               false, false);
}

// Stage one 8192-float (32KB) chunk of packed W into LDS via async loads.
// Each call: 256 threads x b128 = 4KB; 8 calls cover the chunk.
__device__ __forceinline__ void stage_async(float* Wl, const float* Psrc,
                                            int chunk_byte_base, int tid) {
  const unsigned ldsbase = (unsigned)(uintptr_t)Wl;
  const unsigned long long gbase = (unsigned long long)(uintptr_t)Psrc;
#pragma unroll
  for (int i = 0; i < 8; ++i) {
    unsigned dst = ldsbase + (unsigned)(i * 4096 + tid * 16);
    unsigned off = (unsigned)(chunk_byte_base + i * 4096 + tid * 16);
    asm volatile("global_load_async_to_lds_b128 %0, %1, %2"
                 :
                 : "v"(dst), "v"(off), "s"(gbase)
                 : "memory");
  }
  asm volatile("s_wait_asynccnt 0x0" ::: "memory");
}

// Pack W[K,128] (row major) into paired layout:
// P[(k>>1)*256 + col*2 + (k&1)] = W[k*128+col]
__global__ void pack_w_k(const float* __restrict__ W, float* __restrict__ P,
                         int total) {
  const int g = blockIdx.x * blockDim.x + threadIdx.x;
  if (g < total) {
    const int k = g >> 7, col = g & 127;
    P[((k >> 1) << 8) + (col << 1) + (k & 1)] = W[g];
  }
}

// ---------------------------------------------------------------------------
// Generic GEMM: C[M,128] = A[M,K] @ W[K,128] + bias (+ optional ReLU).
// P = packed W. Block = 256 threads = 8 waves; each wave: 32 rows x 128 cols
// (two 16-row tiles sharing B fragments).
// ---------------------------------------------------------------------------
template <int K, bool RELU>
__global__ __launch_bounds__(256) void gemm_n128(const float* __restrict__ P,
                                                 const float* __restrict__ A,
                                                 const float* __restrict__ bias,
                                                 float* __restrict__ C, int M) {
  __shared__ float Wl[64 * 128];  // paired: [kp][col][2], kp in [0,32)
  const int tid  = threadIdx.x;
  const int wave = tid >> 5;
  const int lane = tid & 31;
  const int half = lane >> 4;
  const int lr   = lane & 15;
  const int m0   = blockIdx.x * 256 + wave * 32;

  int r0 = m0 + lr;      if (r0 >= M) r0 = M - 1;
  int r1 = m0 + 16 + lr; if (r1 >= M) r1 = M - 1;
  const float* Ar0 = A + (size_t)r0 * K;
  const float* Ar1 = A + (size_t)r1 * K;

  v8f acc0[8], acc1[8];
#pragma unroll
  for (int t = 0; t < 8; ++t) { acc0[t] = (v8f)0.f; acc1[t] = (v8f)0.f; }

  for (int kc = 0; kc < K; kc += 64) {
    stage_async(Wl, P, kc * 128 * 4, tid);
    __syncthreads();
    __builtin_prefetch(Ar0 + kc + 64, 0, 0);
    __builtin_prefetch(Ar1 + kc + 64, 0, 0);
#pragma unroll
    for (int ks = 0; ks < 64; ks += 4) {
      const float2 a0v = *(const float2*)(Ar0 + kc + ks + 2 * half);
      const float2 a1v = *(const float2*)(Ar1 + kc + ks + 2 * half);
      v2f a0; a0.x = a0v.x; a0.y = a0v.y;
      v2f a1; a1.x = a1v.x; a1.y = a1v.y;
      const float* Wp = Wl + ((ks >> 1) + half) * 256;  // pair row
#pragma unroll
      for (int t = 0; t < 8; ++t) {
        const v2f b = *(const v2f*)(Wp + (t * 16 + lr) * 2);
        acc0[t] = wmma_f32(a0, b, acc0[t]);
        acc1[t] = wmma_f32(a1, b, acc1[t]);
      }
    }
    __syncthreads();
  }

  float bcol[8];
#pragma unroll
  for (int t = 0; t < 8; ++t) bcol[t] = bias[t * 16 + lr];

#pragma unroll
  for (int i = 0; i < 8; ++i) {
    const int rowa = m0 + i + 8 * half;
    if (rowa < M) {
#pragma unroll
      for (int t = 0; t < 8; ++t) {
        float val = acc0[t][i] + bcol[t];
        if (RELU) val = fmaxf(val, 0.f);
        C[(size_t)rowa * 128 + t * 16 + lr] = val;
      }
    }
    const int rowb = m0 + 16 + i + 8 * half;
    if (rowb < M) {
#pragma unroll
      for (int t = 0; t < 8; ++t) {
        float val = acc1[t][i] + bcol[t];
        if (RELU) val = fmaxf(val, 0.f);
        C[(size_t)rowb * 128 + t * 16 + lr] = val;
      }
    }
  }
}

// ---------------------------------------------------------------------------
// Fused incidence GEMM:
//   row r of A = LN(concat(X[v[r]], Xe[e[r]])) * g + b   (stats precomputed)
//   Xev = A @ W2[256,128] + bias, atomically accumulated into Xv_sum[v[r]].
// ---------------------------------------------------------------------------
__global__ __launch_bounds__(256) void conv2_gemm_scatter(
    const float* __restrict__ X, const float* __restrict__ Xe,
    const int* __restrict__ v, const int* __restrict__ e,
    const float* __restrict__ mu, const float* __restrict__ rs,
    const float* __restrict__ g, const float* __restrict__ bln,
    const float* __restrict__ P2, const float* __restrict__ bias,
    float* __restrict__ Xv_sum, int E) {
  __shared__ float Wl[64 * 128];
  const int tid  = threadIdx.x;
  const int wave = tid >> 5;
  const int lane = tid & 31;
  const int half = lane >> 4;
  const int lr   = lane & 15;
  const int m0   = blockIdx.x * 256 + wave * 32;

  int ra = m0 + lr;      if (ra >= E) ra = E - 1;
  int rb = m0 + 16 + lr; if (rb >= E) rb = E - 1;
  const float mu0 = mu[ra], rs0 = rs[ra];
  const float mu1 = mu[rb], rs1 = rs[rb];
  const float* Xr0  = X + (size_t)v[ra] * 128;
  const float* Xer0 = Xe + (size_t)e[ra] * 128;
  const float* Xr1  = X + (size_t)v[rb] * 128;
  const float* Xer1 = Xe + (size_t)e[rb] * 128;

  v8f acc0[8], acc1[8];
#pragma unroll
  for (int t = 0; t < 8; ++t) { acc0[t] = (v8f)0.f; acc1[t] = (v8f)0.f; }

  for (int kc = 0; kc < 256; kc += 64) {
    stage_async(Wl, P2, kc * 128 * 4, tid);
    __syncthreads();
#pragma unroll
    for (int ks = 0; ks < 64; ks += 4) {
      const int c0 = kc + ks + 2 * half;  // even; pair never straddles 128
      float xa0, xa1, xb0, xb1;
      if (c0 < 128) {
        xa0 = Xr0[c0]; xa1 = Xr0[c0 + 1];
        xb0 = Xr1[c0]; xb1 = Xr1[c0 + 1];
      } else {
        xa0 = Xer0[c0 - 128]; xa1 = Xer0[c0 - 127];
        xb0 = Xer1[c0 - 128]; xb1 = Xer1[c0 - 127];
      }
      const float g0 = g[c0], g1 = g[c0 + 1];
      const float l0 = bln[c0], l1 = bln[c0 + 1];
      v2f a0; a0.x = (xa0 - mu0) * rs0 * g0 + l0;
              a0.y = (xa1 - mu0) * rs0 * g1 + l1;
      v2f a1; a1.x = (xb0 - mu1) * rs1 * g0 + l0;
              a1.y = (xb1 - mu1) * rs1 * g1 + l1;
      const float* Wp = Wl + ((ks >> 1) + half) * 256;
#pragma unroll
      for (int t = 0; t < 8; ++t) {
        const v2f b = *(const v2f*)(Wp + (t * 16 + lr) * 2);
        acc0[t] = wmma_f32(a0, b, acc0[t]);
        acc1[t] = wmma_f32(a1, b, acc1[t]);
      }
    }
    __syncthreads();
  }

  float bcol[8];
#pragma unroll
  for (int t = 0; t < 8; ++t) bcol[t] = bias[t * 16 + lr];

#pragma unroll
  for (int i = 0; i < 8; ++i) {
    const int rowa = m0 + i + 8 * half;
    if (rowa < E) {
      float* dst = Xv_sum + (size_t)v[rowa] * 128;
#pragma unroll
      for (int t = 0; t < 8; ++t)
        atomicAdd(dst + t * 16 + lr, acc0[t][i] + bcol[t]);
    }
    const int rowb = m0 + 16 + i + 8 * half;
    if (rowb < E) {
      float* dst = Xv_sum + (size_t)v[rowb] * 128;
#pragma unroll
      for (int t = 0; t < 8; ++t)
        atomicAdd(dst + t * 16 + lr, acc1[t][i] + bcol[t]);
    }
  }
}

// --------------------------- rowwise LayerNorm (H=128) ----------------------
__global__ __launch_bounds__(128) void ln128(const float* __restrict__ Xin,
                                             const float* __restrict__ g,
                                             const float* __restrict__ b,
                                             float* __restrict__ Y) {
  __shared__ float red[128];
  const int row = blockIdx.x;
  const int c   = threadIdx.x;
  const float val = Xin[(size_t)row * 128 + c];
  red[c] = val;
  __syncthreads();
  for (int s = 64; s > 0; s >>= 1) {
    if (c < s) red[c] += red[c + s];
    __syncthreads();
  }
  const float m = red[0] * (1.f / 128.f);
  __syncthreads();
  const float d = val - m;
  red[c] = d * d;
  __syncthreads();
  for (int s = 64; s > 0; s >>= 1) {
    if (c < s) red[c] += red[c + s];
    __syncthreads();
  }
  const float r = rsqrtf(red[0] * (1.f / 128.f) + LNEPS);
  Y[(size_t)row * 128 + c] = d * r * g[c] + b[c];
}

// ------------------ per-incidence LN stats over concat(256) -----------------
__global__ __launch_bounds__(256) void stats256(const float* __restrict__ X,
                                                const float* __restrict__ Xe,
                                                const int* __restrict__ v,
                                                const int* __restrict__ e,
                                                float* __restrict__ mu,
                                                float* __restrict__ rs) {
  __shared__ float s1[256];
  __shared__ float s2[256];
  const int r = blockIdx.x;
  const int c = threadIdx.x;
  const float val = (c < 128) ? X[(size_t)v[r] * 128 + c]
                              : Xe[(size_t)e[r] * 128 + (c - 128)];
  s1[c] = val;
  s2[c] = val * val;
  __syncthreads();
  for (int s = 128; s > 0; s >>= 1) {
    if (c < s) { s1[c] += s1[c + s]; s2[c] += s2[c + s]; }
    __syncthreads();
  }
  if (c == 0) {
    const float m   = s1[0] * (1.f / 256.f);
    const float var = s2[0] * (1.f / 256.f) - m * m;
    mu[r] = m;
    rs[r] = rsqrtf(var + LNEPS);
  }
}

// --------------------------- degree counts ----------------------------------
__global__ void counts_k(const int* __restrict__ v, const int* __restrict__ e,
                         float* __restrict__ cnt_v, float* __restrict__ cnt_e,
                         int E) {
  const int i = blockIdx.x * blockDim.x + threadIdx.x;
  if (i < E) {
    atomicAdd(cnt_v + v[i], 1.f);
    atomicAdd(cnt_e + e[i], 1.f);
  }
}

// ----------------- scatter T[v[i]] into Xe_sum[e[i]] (4 floats/thread) ------
__global__ void scatter_e_k(const float* __restrict__ T,
                            const int* __restrict__ v,
                            const int* __restrict__ e,
                            float* __restrict__ Xe_sum, int E) {
  const int gid = blockIdx.x * blockDim.x + threadIdx.x;
  const int inc = gid >> 5;
  const int c0  = (gid & 31) * 4;
  if (inc < E) {
    const float4 t = *(const float4*)(T + (size_t)v[inc] * 128 + c0);
    float* dst = Xe_sum + (size_t)e[inc] * 128 + c0;
    atomicAdd(dst + 0, t.x);
    atomicAdd(dst + 1, t.y);
    atomicAdd(dst + 2, t.z);
    atomicAdd(dst + 3, t.w);
  }
}

// --------------------------- scatter-mean divide ----------------------------
__global__ void div_mean_k(float* __restrict__ S, const float* __restrict__ cnt,
                           int total) {
  const int g = blockIdx.x * blockDim.x + threadIdx.x;
  if (g < total) {
    const float c = cnt[g >> 7];
    S[g] = (c > 0.f) ? S[g] / c : 0.f;
  }
}

// ------------- X = (1-a)*mean(Xv_sum) + a*X0 (pre-LN combine) ---------------
__global__ void combine_k(const float* __restrict__ Xv_sum,
                          const float* __restrict__ cnt,
                          const float* __restrict__ X0, float* __restrict__ Out,
                          int total) {
  const int g = blockIdx.x * blockDim.x + threadIdx.x;
  if (g < total) {
    const float c  = cnt[g >> 7];
    const float xv = (c > 0.f) ? Xv_sum[g] / c : 0.f;
    Out[g] = (1.f - ALPHAC) * xv + ALPHAC * X0[g];
  }
}

// --------- classifier: relu(X@W1+b1) -> LN(64) -> @W2 + b2 ------------------
__global__ __launch_bounds__(64) void classifier_k(
    const float* __restrict__ X, const float* __restrict__ w1,
    const float* __restrict__ b1, const float* __restrict__ g,
    const float* __restrict__ b, const float* __restrict__ w2,
    const float* __restrict__ b2, float* __restrict__ out) {
  __shared__ float xr[128];
  __shared__ float red[64];
  const int row = blockIdx.x;
  const int j   = threadIdx.x;
  xr[j]      = X[(size_t)row * 128 + j];
  xr[j + 64] = X[(size_t)row * 128 + 64 + j];
  __syncthreads();
  float h = b1[j];
#pragma unroll 8
  for (int k = 0; k < 128; ++k) h += xr[k] * w1[k * 64 + j];
  h = fmaxf(h, 0.f);
  red[j] = h;
  __syncthreads();
  for (int s = 32; s > 0; s >>= 1) {
    if (j < s) red[j] += red[j + s];
    __syncthreads();
  }
  const float m = red[0] * (1.f / 64.f);
  __syncthreads();
  const float d = h - m;
  red[j] = d * d;
  __syncthreads();
  for (int s = 32; s > 0; s >>= 1) {
    if (j < s) red[j] += red[j + s];
    __syncthreads();
  }
  const float r = rsqrtf(red[0] * (1.f / 64.f) + LNEPS);
  __syncthreads();
  red[j] = (d * r * g[j] + b[j]) * w2[j];
  __syncthreads();
  for (int s = 32; s > 0; s >>= 1) {
    if (j < s) red[j] += red[j + s];
    __syncthreads();
  }
  if (j == 0) out[row] = red[0] + b2[0];
}

// ---------------------------------------------------------------------------
extern "C" void kernel_launch(void* const* d_in, const int* in_sizes, int n_in,
                              void* d_out, int out_size, void* d_ws,
                              size_t ws_size, hipStream_t stream) {
  (void)in_sizes; (void)n_in; (void)out_size; (void)ws_size;

  const float* xin[2] = {(const float*)d_in[0], (const float*)d_in[3]};
  const int*   vv[2]  = {(const int*)d_in[1], (const int*)d_in[4]};
  const int*   ee[2]  = {(const int*)d_in[2], (const int*)d_in[5]};
  const float* lin_w  = (const float*)d_in[6];
  const float* lin_b  = (const float*)d_in[7];
  const float* w1_g   = (const float*)d_in[8];
  const float* w1_bln = (const float*)d_in[9];
  const float* w1_w   = (const float*)d_in[10];
  const float* w1_b   = (const float*)d_in[11];
  const float* w2_g   = (const float*)d_in[12];
  const float* w2_bln = (const float*)d_in[13];
  const float* w2_w   = (const float*)d_in[14];
  const float* w2_b   = (const float*)d_in[15];
  const float* w_g    = (const float*)d_in[16];
  const float* w_bln  = (const float*)d_in[17];
  const float* w_w    = (const float*)d_in[18];
  const float* w_b    = (const float*)d_in[19];
  const float* cls_w1 = (const float*)d_in[20];
  const float* cls_b1 = (const float*)d_in[21];
  const float* cls_g  = (const float*)d_in[22];
  const float* cls_b  = (const float*)d_in[23];
  const float* cls_w2 = (const float*)d_in[24];
  const float* cls_b2 = (const float*)d_in[25];

  char* ws = (char*)d_ws;
  size_t off = 0;
  auto alloc = [&](size_t bytes) -> float* {
    float* p = (float*)(ws + off);
    off += (bytes + 255) & ~(size_t)255;
    return p;
  };
  float* X0   = alloc((size_t)N_NODES * 128 * 4);
  float* Xb   = alloc((size_t)N_NODES * 128 * 4);
  float* Y    = alloc((size_t)N_NODES * 128 * 4);
  float* T    = alloc((size_t)N_NODES * 128 * 4);
  float* XeS  = alloc((size_t)N_HEDGES * 128 * 4);
  float* XvS  = alloc((size_t)N_NODES * 128 * 4);
  float* cntE = alloc((size_t)N_HEDGES * 4);
  float* cntV = alloc((size_t)N_NODES * 4);
  float* mu   = alloc((size_t)N_INC * 4);
  float* rs   = alloc((size_t)N_INC * 4);
  float* Plin = alloc((size_t)F_IN * 128 * 4);   // packed lin_w
  float* P1   = alloc((size_t)HID * 128 * 4);    // packed w1_w
  float* P2   = alloc((size_t)256 * 128 * 4);    // packed w2_w
  float* Pw   = alloc((size_t)HID * 128 * 4);    // packed w_w

  // one-shot weight packing (paired layout for ds_load_b64 B fragments)
  pack_w_k<<<(F_IN * 128 + 255) / 256, 256, 0, stream>>>(lin_w, Plin,
                                                         F_IN * 128);
  pack_w_k<<<(HID * 128 + 255) / 256, 256, 0, stream>>>(w1_w, P1, HID * 128);
  pack_w_k<<<(256 * 128 + 255) / 256, 256, 0, stream>>>(w2_w, P2, 256 * 128);
  pack_w_k<<<(HID * 128 + 255) / 256, 256, 0, stream>>>(w_w, Pw, HID * 128);

  const int gRowsN = (N_NODES + 255) / 256;
  const int gRowsE = (N_INC + 255) / 256;

  for (int br = 0; br < 2; ++br) {
    const int* v = vv[br];
    const int* e = ee[br];

    // x0 = relu(x @ lin_w + lin_b)
    gemm_n128<F_IN, true><<<gRowsN, 256, 0, stream>>>(Plin, xin[br], lin_b, X0,
                                                      N_NODES);
    // degree counts (same for both conv iterations)
    hipMemsetAsync(cntE, 0, (size_t)N_HEDGES * 4, stream);
    hipMemsetAsync(cntV, 0, (size_t)N_NODES * 4, stream);
    counts_k<<<(N_INC + 255) / 256, 256, 0, stream>>>(v, e, cntV, cntE, N_INC);

    const float* Xin = X0;
    for (int it = 0; it < 2; ++it) {
      // T = LN(X) @ w1_w + w1_b
      ln128<<<N_NODES, 128, 0, stream>>>(Xin, w1_g, w1_bln, Y);
      gemm_n128<HID, false><<<gRowsN, 256, 0, stream>>>(P1, Y, w1_b, T,
                                                        N_NODES);
      // Xe = scatter_mean(T[v], e)
      hipMemsetAsync(XeS, 0, (size_t)N_HEDGES * 128 * 4, stream);
      scatter_e_k<<<(N_INC * 32 + 255) / 256, 256, 0, stream>>>(T, v, e, XeS,
                                                                N_INC);
      div_mean_k<<<(N_HEDGES * 128 + 255) / 256, 256, 0, stream>>>(
          XeS, cntE, N_HEDGES * 128);
      // per-incidence LN stats of concat(X[v], Xe[e])
      stats256<<<N_INC, 256, 0, stream>>>(Xin, XeS, v, e, mu, rs);
      // Xv_sum += (LN(cat) @ w2_w + w2_b) scattered by v (fused)
      hipMemsetAsync(XvS, 0, (size_t)N_NODES * 128 * 4, stream);
      conv2_gemm_scatter<<<gRowsE, 256, 0, stream>>>(
          Xin, XeS, v, e, mu, rs, w2_g, w2_bln, P2, w2_b, XvS, N_INC);
      // X = relu(LN(0.5*mean(Xv) + 0.5*x0) @ w_w + w_b)
      combine_k<<<(N_NODES * 128 + 255) / 256, 256, 0, stream>>>(
          XvS, cntV, X0, T, N_NODES * 128);
      ln128<<<N_NODES, 128, 0, stream>>>(T, w_g, w_bln, Y);
      gemm_n128<HID, true><<<gRowsN, 256, 0, stream>>>(Pw, Y, w_b, Xb,
                                                       N_NODES);
      Xin = Xb;
    }
    classifier_k<<<N_NODES, 64, 0, stream>>>(
        Xb, cls_w1, cls_b1, cls_g, cls_b, cls_w2, cls_b2,
        (float*)d_out + (size_t)br * N_NODES);
  }
}